// FarmerSimilarityGNN_4140348473961
// MI455X (gfx1250) — compile-verified
//
#include <hip/hip_runtime.h>
#include <math.h>

typedef __attribute__((ext_vector_type(2))) float v2f;
typedef __attribute__((ext_vector_type(8))) float v8f;

#define IN_F   8
#define HIDT   128   // HEADS*HID
#define OUTF   16

__device__ __forceinline__ float lrelu(float x){ return x > 0.f ? x : 0.2f * x; }
__device__ __forceinline__ float elu1 (float x){ return x > 0.f ? x : (expf(x) - 1.f); }
// order-preserving float <-> uint key so we can use integer atomicMax for segment-max
__device__ __forceinline__ unsigned fkey(float x){
  unsigned b = __float_as_uint(x);
  return (b & 0x80000000u) ? ~b : (b | 0x80000000u);
}
__device__ __forceinline__ float funkey(unsigned k){
  return __uint_as_float((k & 0x80000000u) ? (k & 0x7fffffffu) : ~k);
}

__global__ void k_zero(float* __restrict__ p, long long n){
  long long i = blockIdx.x * (long long)blockDim.x + threadIdx.x;
  if (i < n) p[i] = 0.f;   // 0 is also the minimum of the uint key-space (below key(-inf))
}

// ---------------- layer-1 GEMM: h1[N,128] = x[N,8] @ W1[8,128] (WMMA f32 16x16x4) ----
__global__ void k_gemm1(const float* __restrict__ x, const float* __restrict__ W1,
                        float* __restrict__ h1, int n){
  int wave = blockIdx.x * (blockDim.x >> 5) + (threadIdx.x >> 5);
  int lane = threadIdx.x & 31;
  int ntile = (n + 15) >> 4;
  if (wave >= ntile) return;            // wave-uniform: EXEC stays all-ones for WMMA
  int row0 = wave << 4;
  int ln = lane & 15;
  int hi = lane >> 4;                   // half-wave select
  int kg = hi << 1;                     // K offset {0,1} or {2,3} within 4-wide window
  int ar = row0 + ln; if (ar >= n) ar = n - 1;
  bool full = (row0 + 16 <= n);         // wave-uniform fast path (no store guards)
#if __has_builtin(__builtin_amdgcn_wmma_f32_16x16x4_f32)
  v2f a0 = *(const v2f*)(x + (size_t)ar * IN_F + kg);        // K window 0..3
  v2f a1 = *(const v2f*)(x + (size_t)ar * IN_F + 4 + kg);    // K window 4..7
  for (int t = 0; t < 8; ++t){
    int col = t * 16 + ln;
    v2f b0, b1;
    b0[0] = W1[(kg + 0) * HIDT + col]; b0[1] = W1[(kg + 1) * HIDT + col];
    b1[0] = W1[(4 + kg) * HIDT + col]; b1[1] = W1[(5 + kg) * HIDT + col];
    v8f c = {};
    c = __builtin_amdgcn_wmma_f32_16x16x4_f32(false, a0, false, b0, (short)0, c, false, false);
    c = __builtin_amdgcn_wmma_f32_16x16x4_f32(false, a1, false, b1, (short)0, c, false, false);
    if (full){
#pragma unroll
      for (int r = 0; r < 8; ++r)
        h1[(size_t)(row0 + r + (hi << 3)) * HIDT + col] = c[r];
    } else {
#pragma unroll
      for (int r = 0; r < 8; ++r){
        int row = row0 + r + (hi << 3);
        if (row < n) h1[(size_t)row * HIDT + col] = c[r];
      }
    }
  }
#else
  if (hi == 0){
    int row = row0 + ln;
    for (int col = 0; col < HIDT; ++col){
      float acc = 0.f;
      for (int k = 0; k < IN_F; ++k) acc += x[(size_t)ar * IN_F + k] * W1[k * HIDT + col];
      if (row < n) h1[(size_t)row * HIDT + col] = acc;
    }
  }
#endif
}

// ---------------- layer-2 GEMM: t2[N,16] = x2[N,128] @ W2[128,16] --------------------
__global__ void k_gemm2(const float* __restrict__ x2, const float* __restrict__ W2,
                        float* __restrict__ t2, int n){
  int wave = blockIdx.x * (blockDim.x >> 5) + (threadIdx.x >> 5);
  int lane = threadIdx.x & 31;
  int ntile = (n + 15) >> 4;
  if (wave >= ntile) return;
  int row0 = wave << 4;
  int ln = lane & 15;
  int hi = lane >> 4;
  int kg = hi << 1;
  int ar = row0 + ln; if (ar >= n) ar = n - 1;
  bool full = (row0 + 16 <= n);
#if __has_builtin(__builtin_amdgcn_wmma_f32_16x16x4_f32)
  v8f c = {};
  for (int k0 = 0; k0 < HIDT; k0 += 4){
    v2f a = *(const v2f*)(x2 + (size_t)ar * HIDT + k0 + kg);
    v2f b;
    b[0] = W2[(k0 + kg) * OUTF + ln];
    b[1] = W2[(k0 + kg + 1) * OUTF + ln];
    c = __builtin_amdgcn_wmma_f32_16x16x4_f32(false, a, false, b, (short)0, c, false, false);
  }
  if (full){
#pragma unroll
    for (int r = 0; r < 8; ++r)
      t2[(size_t)(row0 + r + (hi << 3)) * OUTF + ln] = c[r];
  } else {
#pragma unroll
    for (int r = 0; r < 8; ++r){
      int row = row0 + r + (hi << 3);
      if (row < n) t2[(size_t)row * OUTF + ln] = c[r];
    }
  }
#else
  if (hi == 0){
    int row = row0 + ln;
    for (int col = 0; col < OUTF; ++col){
      float acc = 0.f;
      for (int k = 0; k < HIDT; ++k) acc += x2[(size_t)ar * HIDT + k] * W2[k * OUTF + col];
      if (row < n) t2[(size_t)row * OUTF + col] = acc;
    }
  }
#endif
}

// ---------------- per-node attention logits -----------------------------------------
__global__ void k_att1(const float* __restrict__ h1, const float* __restrict__ avS,
                       const float* __restrict__ avD, float* __restrict__ as1,
                       float* __restrict__ ad1, int n){
  int i = blockIdx.x * blockDim.x + threadIdx.x;
  if (i >= n) return;
  const float* hr = h1 + (size_t)i * HIDT;
#pragma unroll
  for (int h = 0; h < 4; ++h){
    float s = 0.f, d = 0.f;
#pragma unroll
    for (int j = 0; j < 32; ++j){
      float v = hr[h * 32 + j];
      s += v * avS[h * 32 + j];
      d += v * avD[h * 32 + j];
    }
    as1[(size_t)i * 4 + h] = s;
    ad1[(size_t)i * 4 + h] = d;
  }
}

__global__ void k_att2(const float* __restrict__ t2, const float* __restrict__ avS,
                       const float* __restrict__ avD, float* __restrict__ as2,
                       float* __restrict__ ad2, int n){
  int i = blockIdx.x * blockDim.x + threadIdx.x;
  if (i >= n) return;
  float s = 0.f, d = 0.f;
#pragma unroll
  for (int j = 0; j < OUTF; ++j){
    float v = t2[(size_t)i * OUTF + j];
    s += v * avS[j];
    d += v * avD[j];
  }
  as2[i] = s; ad2[i] = d;
}

// ---------------- edge helpers ------------------------------------------------------
__device__ __forceinline__ void edge_sd(const long long* __restrict__ ei, long long e,
                                        int nE, int* s, int* d){
  if (e < nE){ *s = (int)ei[e]; *d = (int)ei[(size_t)nE + e]; }
  else       { *s = *d = (int)(e - nE); }   // appended self-loops
}

__global__ void k_edge_max1(const long long* __restrict__ ei, const float* __restrict__ as1,
                            const float* __restrict__ ad1, unsigned* __restrict__ m1,
                            int nE, int nN){
  long long e = blockIdx.x * (long long)blockDim.x + threadIdx.x;
  if (e >= (long long)nE + nN) return;
  int s, d; edge_sd(ei, e, nE, &s, &d);
  float4 av = *(const float4*)(as1 + (size_t)s * 4);
  float4 dv = *(const float4*)(ad1 + (size_t)d * 4);
  unsigned* mp = m1 + (size_t)d * 4;
  atomicMax(mp + 0, fkey(lrelu(av.x + dv.x)));
  atomicMax(mp + 1, fkey(lrelu(av.y + dv.y)));
  atomicMax(mp + 2, fkey(lrelu(av.z + dv.z)));
  atomicMax(mp + 3, fkey(lrelu(av.w + dv.w)));
}

__global__ void k_edge_sum1(const long long* __restrict__ ei, const float* __restrict__ as1,
                            const float* __restrict__ ad1, const unsigned* __restrict__ m1,
                            float* __restrict__ dn1, int nE, int nN){
  long long e = blockIdx.x * (long long)blockDim.x + threadIdx.x;
  if (e >= (long long)nE + nN) return;
  int s, d; edge_sd(ei, e, nE, &s, &d);
  float4 av = *(const float4*)(as1 + (size_t)s * 4);
  float4 dv = *(const float4*)(ad1 + (size_t)d * 4);
  uint4 mv = *(const uint4*)(m1 + (size_t)d * 4);
  float* dp = dn1 + (size_t)d * 4;
  atomicAdd(dp + 0, expf(lrelu(av.x + dv.x) - funkey(mv.x)));
  atomicAdd(dp + 1, expf(lrelu(av.y + dv.y) - funkey(mv.y)));
  atomicAdd(dp + 2, expf(lrelu(av.z + dv.z) - funkey(mv.z)));
  atomicAdd(dp + 3, expf(lrelu(av.w + dv.w) - funkey(mv.w)));
}

// one wave per edge: 32 lanes x float4 = 128 accumulated features.
// Edge indices loaded once on lane 0 and broadcast (lane-permute) to save index traffic.
__global__ void k_agg1(const long long* __restrict__ ei, const float* __restrict__ as1,
                       const float* __restrict__ ad1, const unsigned* __restrict__ m1,
                       const float* __restrict__ dn1, const float* __restrict__ h1,
                       float* __restrict__ agg1, int nE, int nN){
  long long gt = blockIdx.x * (long long)blockDim.x + threadIdx.x;
  long long e = gt >> 5;
  int lane = (int)(gt & 31);
  if (e >= (long long)nE + nN) return;
  int s = 0, d = 0;
  if (lane == 0) edge_sd(ei, e, nE, &s, &d);
  s = __shfl(s, 0, 32);
  d = __shfl(d, 0, 32);
  int h = lane >> 3;                        // head for this lane's 4-float chunk
  float el = lrelu(as1[(size_t)s * 4 + h] + ad1[(size_t)d * 4 + h]);
  float alpha = expf(el - funkey(m1[(size_t)d * 4 + h])) / (dn1[(size_t)d * 4 + h] + 1e-16f);
  float4 hv = *(const float4*)(h1 + (size_t)s * HIDT + lane * 4);
  float* out = agg1 + (size_t)d * HIDT + lane * 4;
  atomicAdd(out + 0, hv.x * alpha);
  atomicAdd(out + 1, hv.y * alpha);
  atomicAdd(out + 2, hv.z * alpha);
  atomicAdd(out + 3, hv.w * alpha);
}

__global__ void k_finish1(float* __restrict__ x2, const float* __restrict__ b1, long long len){
  long long i = blockIdx.x * (long long)blockDim.x + threadIdx.x;
  if (i < len) x2[i] = elu1(x2[i] + b1[i & (HIDT - 1)]);
}

__global__ void k_edge_max2(const long long* __restrict__ ei, const float* __restrict__ as2,
                            const float* __restrict__ ad2, unsigned* __restrict__ m2,
                            int nE, int nN){
  long long e = blockIdx.x * (long long)blockDim.x + threadIdx.x;
  if (e >= (long long)nE + nN) return;
  int s, d; edge_sd(ei, e, nE, &s, &d);
  atomicMax(m2 + d, fkey(lrelu(as2[s] + ad2[d])));
}

__global__ void k_edge_sum2(const long long* __restrict__ ei, const float* __restrict__ as2,
                            const float* __restrict__ ad2, const unsigned* __restrict__ m2,
                            float* __restrict__ dn2, int nE, int nN){
  long long e = blockIdx.x * (long long)blockDim.x + threadIdx.x;
  if (e >= (long long)nE + nN) return;
  int s, d; edge_sd(ei, e, nE, &s, &d);
  atomicAdd(dn2 + d, expf(lrelu(as2[s] + ad2[d]) - funkey(m2[d])));
}

// 4 threads per edge: each handles a float4 of the 16 output features
__global__ void k_agg2(const long long* __restrict__ ei, const float* __restrict__ as2,
                       const float* __restrict__ ad2, const unsigned* __restrict__ m2,
                       const float* __restrict__ dn2, const float* __restrict__ t2,
                       float* __restrict__ agg2, int nE, int nN){
  long long gt = blockIdx.x * (long long)blockDim.x + threadIdx.x;
  long long e = gt >> 2;
  int ch = (int)(gt & 3);
  if (e >= (long long)nE + nN) return;
  int s, d; edge_sd(ei, e, nE, &s, &d);
  float alpha = expf(lrelu(as2[s] + ad2[d]) - funkey(m2[d])) / (dn2[d] + 1e-16f);
  float4 hv = *(const float4*)(t2 + (size_t)s * OUTF + ch * 4);
  float* out = agg2 + (size_t)d * OUTF + ch * 4;
  atomicAdd(out + 0, hv.x * alpha);
  atomicAdd(out + 1, hv.y * alpha);
  atomicAdd(out + 2, hv.z * alpha);
  atomicAdd(out + 3, hv.w * alpha);
}

// emb = elu(agg2 + b2); risk = sigmoid(relu(emb@Wc1+bc1)@Wc2+bc2)
__global__ void k_final(const float* __restrict__ agg2, const float* __restrict__ b2,
                        const float* __restrict__ Wc1, const float* __restrict__ bc1,
                        const float* __restrict__ Wc2, const float* __restrict__ bc2,
                        float* __restrict__ out, int n){
  int i = blockIdx.x * blockDim.x + threadIdx.x;
  if (i >= n) return;
  float emb[OUTF];
#pragma unroll
  for (int j = 0; j < OUTF; ++j){
    emb[j] = elu1(agg2[(size_t)i * OUTF + j] + b2[j]);
    out[(size_t)i * OUTF + j] = emb[j];
  }
  float acc = bc2[0];
#pragma unroll
  for (int k = 0; k < 8; ++k){
    float z = bc1[k];
#pragma unroll
    for (int j = 0; j < OUTF; ++j) z += emb[j] * Wc1[j * 8 + k];
    z = z > 0.f ? z : 0.f;
    acc += z * Wc2[k];
  }
  out[(size_t)n * OUTF + i] = 1.f / (1.f + expf(-acc));
}

extern "C" void kernel_launch(void* const* d_in, const int* in_sizes, int n_in,
                              void* d_out, int out_size, void* d_ws, size_t ws_size,
                              hipStream_t stream){
  const float*     x        = (const float*)d_in[0];
  const long long* ei       = (const long long*)d_in[1];   // int64 edge_index [2,E]
  const float*     W1       = (const float*)d_in[2];
  const float*     att_src1 = (const float*)d_in[3];
  const float*     att_dst1 = (const float*)d_in[4];
  const float*     b1       = (const float*)d_in[5];
  const float*     W2       = (const float*)d_in[6];
  const float*     att_src2 = (const float*)d_in[7];
  const float*     att_dst2 = (const float*)d_in[8];
  const float*     b2       = (const float*)d_in[9];
  const float*     Wc1      = (const float*)d_in[10];
  const float*     bc1      = (const float*)d_in[11];
  const float*     Wc2      = (const float*)d_in[12];
  const float*     bc2      = (const float*)d_in[13];

  const int N = in_sizes[0] / IN_F;
  const int E = in_sizes[1] / 2;
  const long long Etot = (long long)E + N;

  // workspace layout (floats/uints, 4B each): 308*N elements total
  float* ws = (float*)d_ws;
  size_t n = (size_t)N;
  float*    h1   = ws;                        // 128N
  float*    as1  = ws + 128 * n;              // 4N
  float*    ad1  = ws + 132 * n;              // 4N
  float*    t2   = ws + 136 * n;              // 16N
  float*    as2  = ws + 152 * n;              // N
  float*    ad2  = ws + 153 * n;              // N
  float*    zb   = ws + 154 * n;              // zero-init region start
  unsigned* m1   = (unsigned*)(ws + 154 * n); // 4N
  float*    dn1  = ws + 158 * n;              // 4N
  float*    x2   = ws + 162 * n;              // 128N (agg1, then elu'd in place)
  unsigned* m2   = (unsigned*)(ws + 290 * n); // N
  float*    dn2  = ws + 291 * n;              // N
  float*    agg2 = ws + 292 * n;              // 16N
  const long long zlen = 154 * (long long)n;

  auto nb = [](long long t, int b){ return (unsigned)((t + b - 1) / b); };
  dim3 B(256);
  int tiles = (N + 15) / 16;

  k_zero     <<<nb(zlen, 256),      B, 0, stream>>>(zb, zlen);
  k_gemm1    <<<nb(tiles, 8),       B, 0, stream>>>(x, W1, h1, N);
  k_att1     <<<nb(N, 256),         B, 0, stream>>>(h1, att_src1, att_dst1, as1, ad1, N);
  k_edge_max1<<<nb(Etot, 256),      B, 0, stream>>>(ei, as1, ad1, m1, E, N);
  k_edge_sum1<<<nb(Etot, 256),      B, 0, stream>>>(ei, as1, ad1, m1, dn1, E, N);
  k_agg1     <<<nb(Etot * 32, 256), B, 0, stream>>>(ei, as1, ad1, m1, dn1, h1, x2, E, N);
  k_finish1  <<<nb(128LL * N, 256), B, 0, stream>>>(x2, b1, 128LL * N);
  k_gemm2    <<<nb(tiles, 8),       B, 0, stream>>>(x2, W2, t2, N);
  k_att2     <<<nb(N, 256),         B, 0, stream>>>(t2, att_src2, att_dst2, as2, ad2, N);
  k_edge_max2<<<nb(Etot, 256),      B, 0, stream>>>(ei, as2, ad2, m2, E, N);
  k_edge_sum2<<<nb(Etot, 256),      B, 0, stream>>>(ei, as2, ad2, m2, dn2, E, N);
  k_agg2     <<<nb(Etot * 4, 256),  B, 0, stream>>>(ei, as2, ad2, m2, dn2, t2, agg2, E, N);
  k_final    <<<nb(N, 256),         B, 0, stream>>>(agg2, b2, Wc1, bc1, Wc2, bc2, (float*)d_out, N);
}